// TorchModel_Rnn_3994319585991
// MI455X (gfx1250) — compile-verified
//
#include <hip/hip_runtime.h>

// ---- CDNA5 wave32 WMMA f32 fragment types ----
typedef __attribute__((ext_vector_type(2))) float v2f;   // A/B frag: 16x4 / 4x16 f32
typedef __attribute__((ext_vector_type(8))) float v8f;   // C/D frag: 16x16 f32

#define BATCH 512
#define SEQ   256
#define HID   512
#define BT    16                 // batch rows per workgroup
#define NW    16                 // waves per workgroup (512 threads, wave32)
#define TPW   2                  // 16-wide column tiles per wave: NW*TPW*16 == HID
#define HSTR  (HID + 4)          // LDS row stride: 516 % 64 == 4 -> conflict-free b64 frag reads

#define WMMA4(A, P, Q, C0, C1)                                                        \
    C0 = __builtin_amdgcn_wmma_f32_16x16x4_f32(false, A, false, P, (short)0, C0, false, false); \
    C1 = __builtin_amdgcn_wmma_f32_16x16x4_f32(false, A, false, Q, (short)0, C1, false, false)

// One 16x16x512 K-sweep feeding TWO output tiles sharing the A fragment.
// 2-stage software pipeline over a ROLLED loop (K loop must stay rolled:
// a full unroll makes the W loads t-invariant and LICM spills them to scratch).
// Tile-1's W row is a constant 16*HID floats past tile-0's -> single base reg.
#define GEMM_2TILE(ABASE, WBASE, C0, C1)                                              \
    do {                                                                              \
        v2f a0 = *(const v2f*)((ABASE) + 0);                                          \
        v2f p0 = *(const v2f*)((WBASE) + 0);                                          \
        v2f q0 = *(const v2f*)((WBASE) + 16 * HID + 0);                               \
        v2f a1 = *(const v2f*)((ABASE) + 4);                                          \
        v2f p1 = *(const v2f*)((WBASE) + 4);                                          \
        v2f q1 = *(const v2f*)((WBASE) + 16 * HID + 4);                               \
        for (int kb = 0; kb < HID - 8; kb += 8) {                                     \
            v2f a2 = *(const v2f*)((ABASE) + kb + 8);                                 \
            v2f p2 = *(const v2f*)((WBASE) + kb + 8);                                 \
            v2f q2 = *(const v2f*)((WBASE) + 16 * HID + kb + 8);                      \
            WMMA4(a0, p0, q0, C0, C1);                                                \
            v2f a3 = *(const v2f*)((ABASE) + kb + 12);                                \
            v2f p3 = *(const v2f*)((WBASE) + kb + 12);                                \
            v2f q3 = *(const v2f*)((WBASE) + 16 * HID + kb + 12);                     \
            WMMA4(a1, p1, q1, C0, C1);                                                \
            a0 = a2; p0 = p2; q0 = q2;                                                \
            a1 = a3; p1 = p3; q1 = q3;                                                \
        }                                                                             \
        WMMA4(a0, p0, q0, C0, C1);                                                    \
        WMMA4(a1, p1, q1, C0, C1);                                                    \
    } while (0)

__global__ __launch_bounds__(512, 1)
void rnn_fused_wmma_kernel(const int* __restrict__ x,        // [B,S]
                           const float* __restrict__ emb,    // [V,H]
                           const float* __restrict__ Wih,    // [H,H]
                           const float* __restrict__ Whh,    // [H,H]
                           const float* __restrict__ Wlin,   // [1,H]
                           const float* __restrict__ blin,   // [1]
                           float* __restrict__ out)          // [B,1]
{
    __shared__ __align__(16) float xs[2][BT][HSTR];  // x_t tiles (ping-pong), 66 KB
    __shared__ __align__(16) float hp[BT][HSTR];     // h_{t-1} tile, 33 KB
    __shared__ float logit[BT];

    const int tid   = threadIdx.x;
    const int wave  = tid >> 5;
    const int lane  = tid & 31;
    const int m     = lane & 15;           // matrix row owned by this lane (A/B frag)
    const int khalf = (lane >> 4) * 2;     // f32 WMMA: lanes 16..31 hold K+2,K+3
    const int rbase = (lane >> 4) * 8;     // C frag: lanes 16..31 hold rows 8..15
    const int b0    = blockIdx.x * BT;
    const int n0    = wave * (TPW * 16);

    // init h_0 = 0, logit accumulators, and stage x_0 (wave `w` gathers emb row `w`)
    for (int i = tid; i < BT * HSTR; i += blockDim.x) ((float*)hp)[i] = 0.0f;
    if (tid < BT) logit[tid] = 0.0f;
    {
        const int tok0 = x[(b0 + wave) * SEQ + 0];
        const float4* src = (const float4*)(emb + (long)tok0 * HID);
        float4* dst = (float4*)(&xs[0][wave][0]);
        #pragma unroll
        for (int j = 0; j < HID / 4 / 32; ++j) dst[lane + 32 * j] = src[lane + 32 * j];
    }
    __syncthreads();

    v8f acc_sum[TPW];
    #pragma unroll
    for (int tt = 0; tt < TPW; ++tt) acc_sum[tt] = v8f{0.f,0.f,0.f,0.f,0.f,0.f,0.f,0.f};

    // single base pointer per weight matrix; tile-1 row handled by +16*HID immediate
    const float* wi = Wih + (long)(n0 + m) * HID + khalf;
    const float* wh = Whh + (long)(n0 + m) * HID + khalf;

    int p = 0;
    for (int t = 0; t < SEQ; ++t) {
        // Issue next step's embedding gather NOW; commit to LDS after GEMM 1.
        float4 stg[HID / 4 / 32];
        if (t + 1 < SEQ) {
            const int tokn = x[(b0 + wave) * SEQ + (t + 1)];
            const float4* src = (const float4*)(emb + (long)tokn * HID);
            #pragma unroll
            for (int j = 0; j < HID / 4 / 32; ++j) stg[j] = src[lane + 32 * j];
        }

        v8f acc0 = v8f{0.f,0.f,0.f,0.f,0.f,0.f,0.f,0.f};
        v8f acc1 = v8f{0.f,0.f,0.f,0.f,0.f,0.f,0.f,0.f};

        // ---- x_t @ Wih^T : A from LDS (staged gather), B streamed from L2 ----
        GEMM_2TILE(&xs[p][m][khalf], wi, acc0, acc1);

        // commit staged x_{t+1} (gather loads had all of GEMM-1 to land; frees regs)
        if (t + 1 < SEQ) {
            float4* dst = (float4*)(&xs[p ^ 1][wave][0]);
            #pragma unroll
            for (int j = 0; j < HID / 4 / 32; ++j) dst[lane + 32 * j] = stg[j];
        }

        // ---- h_{t-1} @ Whh^T : A from LDS, B streamed from L2 ----
        GEMM_2TILE(&hp[m][khalf], wh, acc0, acc1);

        __syncthreads();   // all waves done READING hp / xs[p]

        // tanh + write h_t (wave owns columns [n0, n0+32)) + pooled running sum
        #pragma unroll
        for (int tt = 0; tt < TPW; ++tt) {
            const int n = n0 + tt * 16 + m;
            v8f& a = (tt == 0) ? acc0 : acc1;
            #pragma unroll
            for (int r = 0; r < 8; ++r) {
                const float hv = tanhf(a[r]);
                hp[rbase + r][n] = hv;
                acc_sum[tt][r] += hv;
            }
        }
        __syncthreads();   // h_t and x_{t+1} visible to all waves
        p ^= 1;
    }

    // ---- pooled @ Wlin^T + b, sigmoid ----
    #pragma unroll
    for (int tt = 0; tt < TPW; ++tt) {
        const int n = n0 + tt * 16 + m;
        const float wl = Wlin[n];
        #pragma unroll
        for (int r = 0; r < 8; ++r) {
            atomicAdd(&logit[rbase + r], acc_sum[tt][r] * wl);   // ds_add_f32
        }
    }
    __syncthreads();
    if (tid < BT) {
        const float z = logit[tid] * (1.0f / (float)SEQ) + blin[0];
        out[b0 + tid] = 1.0f / (1.0f + __expf(-z));
    }
}

extern "C" void kernel_launch(void* const* d_in, const int* in_sizes, int n_in,
                              void* d_out, int out_size, void* d_ws, size_t ws_size,
                              hipStream_t stream) {
    (void)in_sizes; (void)n_in; (void)d_ws; (void)ws_size; (void)out_size;
    const int*   x    = (const int*)  d_in[0];
    const float* emb  = (const float*)d_in[1];
    const float* Wih  = (const float*)d_in[2];
    const float* Whh  = (const float*)d_in[3];
    const float* Wlin = (const float*)d_in[4];
    const float* blin = (const float*)d_in[5];
    float* out = (float*)d_out;

    dim3 grid(BATCH / BT);   // 32 persistent workgroups
    dim3 block(NW * 32);     // 512 threads = 16 wave32s
    rnn_fused_wmma_kernel<<<grid, block, 0, stream>>>(x, emb, Wih, Whh, Wlin, blin, out);
}